// MVProp2D_70531952935306
// MI455X (gfx1250) — compile-verified
//
#include <hip/hip_runtime.h>
#include <hip/hip_bf16.h>
#include <math.h>

typedef __attribute__((ext_vector_type(2))) float v2f;
typedef __attribute__((ext_vector_type(8))) float v8f;

__device__ __forceinline__ float sigmoidf_fast(float x) {
    return 1.0f / (1.0f + __expf(-x));
}

// -----------------------------------------------------------------------------
// Kernel 1: gather_map = sigmoid(image[p,:] . gather_w + gather_b)
// WMMA f32 16x16x4, 16 pixels per wave, K=64 via 16 chained WMMAs.
// gather_w broadcast to all 16 N-columns of B => every column of D = dot(pixel).
// -----------------------------------------------------------------------------
__global__ void __launch_bounds__(256)
gather_wmma_kernel(const float* __restrict__ image,
                   const float* __restrict__ gather_w,
                   const float* __restrict__ gather_b,
                   float* __restrict__ out_gm) {
    __shared__ float wlds[64];
    if (threadIdx.x < 64) wlds[threadIdx.x] = gather_w[threadIdx.x];
    __syncthreads();

    const int lane = threadIdx.x & 31;
    const int wave = threadIdx.x >> 5;
    const size_t tile    = (size_t)blockIdx.x * 8 + wave;   // 65536 tiles total
    const size_t pixBase = tile * 16;                       // 16 pixels / tile
    const int m  = lane & 15;                // pixel row within tile (A: M)
    const int kh = (lane >> 4) << 1;         // K sub-offset: 0 (lanes 0-15) or 2

    const float* row = image + (pixBase + (size_t)m) * 64;

    v8f acc = {};
#pragma unroll
    for (int k0 = 0; k0 < 64; k0 += 4) {
        v2f a, b;
        a.x = row[k0 + kh];
        a.y = row[k0 + kh + 1];
        b.x = wlds[k0 + kh];        // same value for every N column
        b.y = wlds[k0 + kh + 1];
        acc = __builtin_amdgcn_wmma_f32_16x16x4_f32(
            /*neg_a=*/false, a, /*neg_b=*/false, b,
            /*c_mod=*/(short)0, acc, /*reuse_a=*/false, /*reuse_b=*/false);
    }

    // D layout: lanes 0-15 VGPR r -> M=r ; lanes 16-31 VGPR r -> M=r+8.
    // All N columns identical: lane L(0..7) emits pixel L, lane 16+L emits 8+L.
    const int s = lane & 7;
    float val = acc[0];
    if (s == 1) val = acc[1];
    if (s == 2) val = acc[2];
    if (s == 3) val = acc[3];
    if (s == 4) val = acc[4];
    if (s == 5) val = acc[5];
    if (s == 6) val = acc[6];
    if (s == 7) val = acc[7];

    if ((lane & 15) < 8) {
        const size_t pix = pixBase + ((size_t)(lane >> 4) << 3) + s;
        out_gm[pix] = sigmoidf_fast(val + gather_b[0]);
    }
}

// -----------------------------------------------------------------------------
// Kernel 2: reward_locations = minmax_norm(relu(rewarding_states - bias))
// One block per batch image (256x256 = 65536 elems), LDS tree reduction.
// -----------------------------------------------------------------------------
__global__ void __launch_bounds__(256)
reward_norm_kernel(const float* __restrict__ rs,
                   const float* __restrict__ adjust_bias,
                   float* __restrict__ out_rl) {
    const int b   = blockIdx.x;
    const int tid = threadIdx.x;
    const int N   = 256 * 256;
    const float bias = adjust_bias[0];
    const float* x = rs + (size_t)b * N;

    float mn =  INFINITY;
    float mx = -INFINITY;
    for (int i = tid; i < N; i += 256) {
        float v = fmaxf(x[i] - bias, 0.0f);
        mn = fminf(mn, v);
        mx = fmaxf(mx, v);
    }
    __shared__ float smn[256], smx[256];
    smn[tid] = mn; smx[tid] = mx;
    __syncthreads();
    for (int stp = 128; stp > 0; stp >>= 1) {
        if (tid < stp) {
            smn[tid] = fminf(smn[tid], smn[tid + stp]);
            smx[tid] = fmaxf(smx[tid], smx[tid + stp]);
        }
        __syncthreads();
    }
    mn = smn[0]; mx = smx[0];
    const float inv = 1.0f / (mx - mn + 0.0001f);

    float* o = out_rl + (size_t)b * N;
    for (int i = tid; i < N; i += 256) {
        float v = fmaxf(x[i] - bias, 0.0f);
        o[i] = (v - mn) * inv;
    }
}

// -----------------------------------------------------------------------------
// Kernel 3: 10 iterations of vm = max(gm * weighted 3x3 neighbor max, vm),
// then value_map = relu(vm - reward_locations).
// Tiled: 32x32 output + 10-halo = 52x52 in LDS, shrinking valid region, no
// global sync between iterations. Out-of-image cells are 0 and stay 0
// (gm=0 there), matching per-iteration zero padding in the reference.
// -----------------------------------------------------------------------------
#define TS   32
#define HALO 10
#define S    (TS + 2 * HALO)     // 52
#define SP   (S + 1)             // 53: padded row stride

__global__ void __launch_bounds__(256)
propagate_kernel(const float* __restrict__ rl,   // reward_locations (out slice 2)
                 const float* __restrict__ gm,   // gather_map       (out slice 0)
                 const float* __restrict__ gamma_weight,
                 float* __restrict__ out_vmap) {
    __shared__ float bufA[S * SP];
    __shared__ float bufB[S * SP];
    __shared__ float gms [S * SP];

    const int tid = threadIdx.x;
    const int b   = blockIdx.z;
    const int ty0 = blockIdx.y * TS;
    const int tx0 = blockIdx.x * TS;

    // weights = sigmoid(gamma_weight) * sigmoid(VALUE_FLOW_LOGITS)
    const float gamma = sigmoidf_fast(gamma_weight[0]);
    const float wHi = gamma * sigmoidf_fast( 10.0f);  // plus-shape + center
    const float wLo = gamma * sigmoidf_fast(-10.0f);  // corners
    const float w00 = wLo, w01 = wHi, w02 = wLo;
    const float w10 = wHi, w11 = wHi, w12 = wHi;
    const float w20 = wLo, w21 = wHi, w22 = wLo;

    // Load 52x52 tile (vm initial = reward_locations; gather map) with 0 halo.
    for (int idx = tid; idx < S * S; idx += 256) {
        const int i = idx / S, j = idx - i * S;
        const int gy = ty0 - HALO + i;
        const int gx = tx0 - HALO + j;
        float v = 0.0f, g = 0.0f;
        if (gy >= 0 && gy < 256 && gx >= 0 && gx < 256) {
            const size_t off = ((size_t)b * 256 + gy) * 256 + gx;
            v = rl[off];
            g = gm[off];
        }
        bufA[i * SP + j] = v;
        gms [i * SP + j] = g;
    }
    __syncthreads();

    float* cur = bufA;
    float* nxt = bufB;
#pragma unroll 1
    for (int t = 0; t < 10; ++t) {
        for (int idx = tid; idx < S * S; idx += 256) {
            const int i = idx / S, j = idx - i * S;
            const float c = cur[i * SP + j];
            float r = c;
            if (i >= 1 && i < S - 1 && j >= 1 && j < S - 1) {
                const float* p0 = cur + (i - 1) * SP + (j - 1);
                const float* p1 = cur + (i    ) * SP + (j - 1);
                const float* p2 = cur + (i + 1) * SP + (j - 1);
                float mx;
                mx =            p0[0] * w00;
                mx = fmaxf(mx,  p0[1] * w01);
                mx = fmaxf(mx,  p0[2] * w02);
                mx = fmaxf(mx,  p1[0] * w10);
                mx = fmaxf(mx,  p1[1] * w11);
                mx = fmaxf(mx,  p1[2] * w12);
                mx = fmaxf(mx,  p2[0] * w20);
                mx = fmaxf(mx,  p2[1] * w21);
                mx = fmaxf(mx,  p2[2] * w22);
                r = fmaxf(gms[i * SP + j] * mx, c);
            }
            nxt[i * SP + j] = r;
        }
        __syncthreads();
        float* tmp = cur; cur = nxt; nxt = tmp;
    }

    // value_map = relu(vm - reward_locations) on the inner 32x32.
    for (int idx = tid; idx < TS * TS; idx += 256) {
        const int oy = idx / TS, ox = idx - oy * TS;
        const int i  = oy + HALO, j = ox + HALO;
        const size_t off = ((size_t)b * 256 + (ty0 + oy)) * 256 + (tx0 + ox);
        out_vmap[off] = fmaxf(cur[i * SP + j] - rl[off], 0.0f);
    }
}

// -----------------------------------------------------------------------------
// Launch
// inputs: image(16*256*256*64 f32), rewarding_states(16*256*256 f32),
//         gather_w(64 f32), gather_b(1), gamma_weight(1), adjust_bias(1)
// outputs concat: gather_map | value_map | reward_locations, each 1048576 f32
// -----------------------------------------------------------------------------
extern "C" void kernel_launch(void* const* d_in, const int* in_sizes, int n_in,
                              void* d_out, int out_size, void* d_ws, size_t ws_size,
                              hipStream_t stream) {
    const float* image        = (const float*)d_in[0];
    const float* rs           = (const float*)d_in[1];
    const float* gather_w     = (const float*)d_in[2];
    const float* gather_b     = (const float*)d_in[3];
    const float* gamma_weight = (const float*)d_in[4];
    const float* adjust_bias  = (const float*)d_in[5];

    const size_t NPIX = (size_t)16 * 256 * 256;        // 1,048,576
    float* out_gm   = (float*)d_out;                   // gather_map
    float* out_vmap = (float*)d_out + NPIX;            // value_map
    float* out_rl   = (float*)d_out + 2 * NPIX;        // reward_locations

    // 1) gather_map via WMMA: 65536 tiles of 16 pixels, 8 waves/block.
    gather_wmma_kernel<<<8192, 256, 0, stream>>>(image, gather_w, gather_b, out_gm);

    // 2) reward_locations: one block per batch.
    reward_norm_kernel<<<16, 256, 0, stream>>>(rs, adjust_bias, out_rl);

    // 3) propagation + value_map: 8x8 tiles x 16 batches.
    dim3 grid(256 / TS, 256 / TS, 16);
    propagate_kernel<<<grid, 256, 0, stream>>>(out_rl, out_gm, gamma_weight, out_vmap);
}